// RFA_30485677867995
// MI455X (gfx1250) — compile-verified
//
#include <hip/hip_runtime.h>
#include <cmath>

// ---------------------------------------------------------------------------
// RFA (random feature attention), fp32, CDNA5/gfx1250.
// All GEMM-shaped work uses V_WMMA_F32_16X16X4_F32, fed from LDS with
// double-buffered staging. Staging uses GLOBAL_LOAD_ASYNC_TO_LDS_B128
// (ASYNCcnt) when the toolchain exposes it, else VGPR round-trip.
// Causal recurrence is chunked linear attention (T=64).
// ---------------------------------------------------------------------------

typedef float v2f __attribute__((ext_vector_type(2)));
typedef float v8f __attribute__((ext_vector_type(8)));
typedef int v4i __attribute__((ext_vector_type(4)));

#define BSZ 2
#define LSEQ 1024
#define DIMM 1024
#define NH 16
#define NTOK (BSZ * LSEQ)
#define MIN_LOG_STD (-11.512925464970229f) /* log(1e-5) */

#if defined(__has_builtin)
#if __has_builtin(__builtin_amdgcn_global_load_async_to_lds_b128)
#define ASYNC_LDS 1
#endif
#endif

// Copy 16 bytes global -> LDS (async direct-to-LDS when available).
static __device__ __forceinline__ void cp_b128(const float* g, float* l) {
#ifdef ASYNC_LDS
  __builtin_amdgcn_global_load_async_to_lds_b128(
      (__attribute__((address_space(1))) v4i*)(float*)g,
      (__attribute__((address_space(3))) v4i*)l, 0, 0);
#else
  *(float4*)l = *(const float4*)g;
#endif
}

// Wait for this wave's outstanding async LDS copies (no-op on fallback path).
static __device__ __forceinline__ void async_wait() {
#ifdef ASYNC_LDS
#if defined(__has_builtin) && __has_builtin(__builtin_amdgcn_s_wait_asynccnt)
  __builtin_amdgcn_s_wait_asynccnt(0);
#else
  asm volatile("s_wait_asynccnt 0x0" ::: "memory");
#endif
#endif
}

static __device__ __forceinline__ v8f wmma4(v2f a, v2f b, v8f c) {
  // D = A(16x4 f32) x B(4x16 f32) + C(16x16 f32)
  return __builtin_amdgcn_wmma_f32_16x16x4_f32(false, a, false, b, (short)0, c,
                                               false, false);
}

static __device__ __forceinline__ v8f zero8() {
  v8f z = {0.f, 0.f, 0.f, 0.f, 0.f, 0.f, 0.f, 0.f};
  return z;
}

// ---------------------------------------------------------------------------
// C[m,n] = sum_k A[m,k] * B[n,k] + bias[n]   (A: M x K, B: N x K, row major)
// Block tile 128(M) x 128(N); K staged in slabs of 16 through double-buffered
// LDS (2 x (8KB A + 8KB B)). 8 waves; wave w computes rows w*16..w*16+15 x
// all 128 columns (8 f32-WMMA accumulators). Grid: (N/128, M/128).
// ---------------------------------------------------------------------------
__global__ __launch_bounds__(256) void gemm_absT(const float* __restrict__ A,
                                                 const float* __restrict__ B,
                                                 const float* __restrict__ bias,
                                                 float* __restrict__ C,
                                                 int K, int ldc) {
  __shared__ __align__(16) float As[2][128 * 16]; // [row][k]
  __shared__ __align__(16) float Bs[2][128 * 16]; // [col][k]

  const int tid = threadIdx.x;
  const int lane = tid & 31;
  const int wave = tid >> 5;
  const int l16 = lane & 15;
  const int kk = (lane >> 4) << 1;
  const int mblk = blockIdx.y * 128;
  const int nblk = blockIdx.x * 128;

  // Staging map: 512 float4 per tile; thread handles f4-index tid and tid+256.
  const int r0 = tid >> 2, c0 = (tid & 3) << 2;
  const int r1 = (tid + 256) >> 2, c1 = ((tid + 256) & 3) << 2;

  const float* Ar0 = A + (size_t)(mblk + r0) * K + c0;
  const float* Ar1 = A + (size_t)(mblk + r1) * K + c1;
  const float* Br0 = B + (size_t)(nblk + r0) * K + c0;
  const float* Br1 = B + (size_t)(nblk + r1) * K + c1;

  v8f acc[8];
#pragma unroll
  for (int j = 0; j < 8; ++j) acc[j] = zero8();

  // prologue: stage slab k0=0 into buffer 0
  cp_b128(Ar0, &As[0][r0 * 16 + c0]);
  cp_b128(Ar1, &As[0][r1 * 16 + c1]);
  cp_b128(Br0, &Bs[0][r0 * 16 + c0]);
  cp_b128(Br1, &Bs[0][r1 * 16 + c1]);
  async_wait();
  __syncthreads();

  int buf = 0;
  for (int k0 = 0; k0 < K; k0 += 16) {
    // kick off next slab's copies into the other buffer (overlaps compute)
    const bool havenext = (k0 + 16) < K;
    if (havenext) {
      cp_b128(Ar0 + k0 + 16, &As[buf ^ 1][r0 * 16 + c0]);
      cp_b128(Ar1 + k0 + 16, &As[buf ^ 1][r1 * 16 + c1]);
      cp_b128(Br0 + k0 + 16, &Bs[buf ^ 1][r0 * 16 + c0]);
      cp_b128(Br1 + k0 + 16, &Bs[buf ^ 1][r1 * 16 + c1]);
    }

    // compute on current slab from LDS
    const float* Ab = As[buf];
    const float* Bb = Bs[buf];
#pragma unroll
    for (int ks = 0; ks < 4; ++ks) {
      v2f a = *(const v2f*)(Ab + (wave * 16 + l16) * 16 + ks * 4 + kk);
#pragma unroll
      for (int j = 0; j < 8; ++j) {
        v2f b = *(const v2f*)(Bb + (j * 16 + l16) * 16 + ks * 4 + kk);
        acc[j] = wmma4(a, b, acc[j]);
      }
    }

    async_wait();
    __syncthreads();
    buf ^= 1;
  }

  const int mtop = mblk + wave * 16 + ((lane >> 4) << 3);
#pragma unroll
  for (int j = 0; j < 8; ++j) {
    const int n = nblk + j * 16 + l16;
    const float bs = bias[n];
#pragma unroll
    for (int i = 0; i < 8; ++i)
      C[(size_t)(mtop + i) * ldc + n] = acc[j][i] + bs;
  }
}

// ---------------------------------------------------------------------------
// rp[h,p,d] = exp(max(log_sigma[h,0,d], MIN_LOG_STD)) * random_projs[0,p,d]
// ---------------------------------------------------------------------------
__global__ __launch_bounds__(256) void build_rp(const float* __restrict__ log_sigma,
                                                const float* __restrict__ rproj,
                                                float* __restrict__ rp) {
  const int idx = blockIdx.x * 256 + threadIdx.x; // < 16*64*64
  const int d = idx & 63;
  const int h = idx >> 12;
  const int pd = idx & 4095;
  const float ls = log_sigma[(h << 6) + d];
  rp[idx] = __expf(fmaxf(ls, MIN_LOG_STD)) * rproj[pd];
}

// ---------------------------------------------------------------------------
// phi = relu(normalize(Q) @ rp[h]^T) / sqrt(64) [* mask], in place.
// Block: 128 tokens x 1 head. Stage X slab (128x64) + rp[h] (64x64) in LDS,
// fuse L2 row normalization, then pure LDS-fed WMMA. grid = (tokens/128, H).
// ---------------------------------------------------------------------------
__global__ __launch_bounds__(256) void phi_kernel(float* __restrict__ Q,
                                                  const float* __restrict__ rp,
                                                  const unsigned char* __restrict__ mask,
                                                  int applyMask) {
  __shared__ __align__(16) float Xs[128 * 64]; // 32KB
  __shared__ __align__(16) float Rs[64 * 64];  // 16KB

  const int tid = threadIdx.x;
  const int lane = tid & 31;
  const int wave = tid >> 5;
  const int l16 = lane & 15;
  const int kk = (lane >> 4) << 1;
  const int head = blockIdx.y;
  const int tokbase = blockIdx.x * 128;

  // stage X: 128 rows of 64 (2048 float4, 8 per thread)
  for (int i = tid; i < 2048; i += 256) {
    const int row = i >> 4, f4 = (i & 15) << 2;
    cp_b128(Q + (size_t)(tokbase + row) * DIMM + head * 64 + f4,
            &Xs[row * 64 + f4]);
  }
  // stage rp[h]: 64x64 (1024 float4, 4 per thread)
  for (int i = tid; i < 1024; i += 256) {
    const int f4 = i << 2;
    cp_b128(rp + head * 4096 + f4, &Rs[f4]);
  }
  async_wait();
  __syncthreads();

  // fused L2 normalization of each token row (in LDS)
  if (tid < 128) {
    float ss = 0.f;
#pragma unroll
    for (int d = 0; d < 64; ++d) {
      const float t = Xs[tid * 64 + d];
      ss += t * t;
    }
    const float inv = 1.f / fmaxf(sqrtf(ss), 1e-12f);
#pragma unroll
    for (int d = 0; d < 64; ++d) Xs[tid * 64 + d] *= inv;
  }
  __syncthreads();

  v8f acc[4];
#pragma unroll
  for (int j = 0; j < 4; ++j) acc[j] = zero8();

  const int arow = wave * 16 + l16;
  for (int k0 = 0; k0 < 64; k0 += 4) {
    v2f a = *(const v2f*)(Xs + arow * 64 + k0 + kk);
#pragma unroll
    for (int j = 0; j < 4; ++j) {
      v2f b = *(const v2f*)(Rs + (j * 16 + l16) * 64 + k0 + kk);
      acc[j] = wmma4(a, b, acc[j]);
    }
  }

  const int mtop = tokbase + wave * 16 + ((lane >> 4) << 3);
#pragma unroll
  for (int i = 0; i < 8; ++i) {
    const int tok = mtop + i;
    const float mval = applyMask ? (mask[tok] ? 0.125f : 0.f) : 0.125f;
#pragma unroll
    for (int j = 0; j < 4; ++j) {
      const int p = j * 16 + l16;
      Q[(size_t)tok * DIMM + head * 64 + p] = fmaxf(acc[j][i], 0.f) * mval;
    }
  }
}

// ---------------------------------------------------------------------------
// Chunked causal RFA. One workgroup per (head, batch); 8 waves, T = 64.
//   Am  = causal_mask(Phi_q Phi_k^T)          (64x64, LDS)
//   den = rowsum(Am) + Phi_q . z
//   num = Am @ V + Phi_q @ S ; out = num/den
//   S  += Phi_k^T @ V ; z += colsum(Phi_k)
// S, Qc, Kc, Vc, Am all in (dynamic) LDS: 80.5 KB.
// ---------------------------------------------------------------------------
__global__ __launch_bounds__(256) void rfa_kernel(const float* __restrict__ phq,
                                                  const float* __restrict__ phk,
                                                  const float* __restrict__ vv,
                                                  float* __restrict__ hb) {
  extern __shared__ float smem[];
  float* S = smem;            // 4096
  float* Qc = S + 4096;       // 4096
  float* Kc = Qc + 4096;      // 4096
  float* Vc = Kc + 4096;      // 4096
  float* Am = Vc + 4096;      // 4096
  float* zv = Am + 4096;      // 64
  float* den = zv + 64;       // 64

  const int head = blockIdx.x;
  const int b = blockIdx.y;
  const int tid = threadIdx.x;
  const int lane = tid & 31;
  const int wave = tid >> 5;
  const int l16 = lane & 15;
  const int kk = (lane >> 4) << 1;

  for (int i = tid; i < 4096; i += 256) S[i] = 0.f;
  if (tid < 64) zv[tid] = 0.f;
  __syncthreads();

  for (int c0 = 0; c0 < LSEQ; c0 += 64) {
    const size_t base = ((size_t)(b * LSEQ + c0) * NH + head) * 64;
    const float* pqc = phq + base; // [t][d], row stride DIMM
    const float* pkc = phk + base;
    const float* vc = vv + base;

    // stage Qc, Kc, Vc (each 64x64: 1024 float4, 4 per thread)
    for (int i = tid; i < 1024; i += 256) {
      const int t = i >> 4, f4 = (i & 15) << 2;
      const size_t g = (size_t)t * DIMM + f4;
      cp_b128(pqc + g, &Qc[t * 64 + f4]);
      cp_b128(pkc + g, &Kc[t * 64 + f4]);
      cp_b128(vc + g, &Vc[t * 64 + f4]);
    }
    async_wait();
    __syncthreads();

    // ---- scores: Am = causal_mask(Qc @ Kc^T), 16 tiles, 2 per wave -------
#pragma unroll
    for (int u = 0; u < 2; ++u) {
      const int tile = wave * 2 + u;
      const int tm = (tile >> 2) << 4, tn = (tile & 3) << 4;
      v8f acc = zero8();
      const int arow = tm + l16;
      for (int k0 = 0; k0 < 64; k0 += 4) {
        v2f a = *(const v2f*)(Qc + arow * 64 + k0 + kk);
        v2f bfr = *(const v2f*)(Kc + (tn + l16) * 64 + k0 + kk);
        acc = wmma4(a, bfr, acc);
      }
      const int m0 = tm + ((lane >> 4) << 3);
      const int n = tn + l16;
#pragma unroll
      for (int i = 0; i < 8; ++i) {
        const int t = m0 + i;
        Am[t * 64 + n] = (t >= n) ? acc[i] : 0.f;
      }
    }
    __syncthreads();

    // ---- denominator -----------------------------------------------------
    if (tid < 64) {
      float s = 0.f;
      for (int j = 0; j < 64; ++j) s += Am[tid * 64 + j];
      for (int p = 0; p < 64; ++p) s += Qc[tid * 64 + p] * zv[p];
      den[tid] = fmaxf(s, 1e-5f);
    }
    __syncthreads();

    // ---- numerator: Am @ V + Qc @ S_prev, then out = num/den -------------
#pragma unroll
    for (int u = 0; u < 2; ++u) {
      const int tile = wave * 2 + u;
      const int tm = (tile >> 2) << 4, tn = (tile & 3) << 4;
      v8f acc = zero8();
      const int arow = tm + l16;
      const int n = tn + l16;
      for (int k0 = 0; k0 < 64; k0 += 4) {
        v2f a = *(const v2f*)(Am + arow * 64 + k0 + kk);
        v2f bfr;
        bfr.x = Vc[(k0 + kk) * 64 + n];
        bfr.y = Vc[(k0 + kk + 1) * 64 + n];
        acc = wmma4(a, bfr, acc);
      }
      for (int k0 = 0; k0 < 64; k0 += 4) {
        v2f a = *(const v2f*)(Qc + arow * 64 + k0 + kk);
        v2f bfr;
        bfr.x = S[(k0 + kk) * 64 + n];
        bfr.y = S[(k0 + kk + 1) * 64 + n];
        acc = wmma4(a, bfr, acc);
      }
      const int m0 = tm + ((lane >> 4) << 3);
#pragma unroll
      for (int i = 0; i < 8; ++i) {
        const int t = m0 + i;
        hb[(size_t)(b * LSEQ + c0 + t) * DIMM + head * 64 + n] = acc[i] / den[t];
      }
    }
    __syncthreads();

    // ---- state update: S += Kc^T @ V ; z += colsum(Kc) -------------------
#pragma unroll
    for (int u = 0; u < 2; ++u) {
      const int tile = wave * 2 + u;
      const int tm = (tile >> 2) << 4, tn = (tile & 3) << 4; // (p, d) tile
      const int m0 = tm + ((lane >> 4) << 3);
      const int n = tn + l16;
      v8f acc;
#pragma unroll
      for (int i = 0; i < 8; ++i) acc[i] = S[(m0 + i) * 64 + n];
      const int arow = tm + l16;
      for (int k0 = 0; k0 < 64; k0 += 4) {
        v2f a, bfr;
        a.x = Kc[(k0 + kk) * 64 + arow];
        a.y = Kc[(k0 + kk + 1) * 64 + arow];
        bfr.x = Vc[(k0 + kk) * 64 + n];
        bfr.y = Vc[(k0 + kk + 1) * 64 + n];
        acc = wmma4(a, bfr, acc);
      }
#pragma unroll
      for (int i = 0; i < 8; ++i) S[(m0 + i) * 64 + n] = acc[i];
    }
    if (tid < 64) {
      float s = zv[tid];
      for (int t = 0; t < 64; ++t) s += Kc[t * 64 + tid];
      zv[tid] = s;
    }
    __syncthreads();
  }
}

// ---------------------------------------------------------------------------
extern "C" void kernel_launch(void* const* d_in, const int* in_sizes, int n_in,
                              void* d_out, int out_size, void* d_ws, size_t ws_size,
                              hipStream_t stream) {
  const float* x = (const float*)d_in[0];
  const unsigned char* mask = (const unsigned char*)d_in[1]; // bool_
  const float* Wq = (const float*)d_in[2];
  const float* bq = (const float*)d_in[3];
  const float* Wk = (const float*)d_in[4];
  const float* bk = (const float*)d_in[5];
  const float* Wv = (const float*)d_in[6];
  const float* bv = (const float*)d_in[7];
  const float* Wo = (const float*)d_in[8];
  const float* bo = (const float*)d_in[9];
  const float* log_sigma = (const float*)d_in[10];
  const float* rproj = (const float*)d_in[11];

  // workspace: q | k | v | h | rp  (4*8MB + 256KB ~= 34 MB)
  float* ws = (float*)d_ws;
  float* q = ws;                       // becomes phi_q in place
  float* k = q + (size_t)NTOK * DIMM;  // becomes phi_k in place
  float* v = k + (size_t)NTOK * DIMM;
  float* hb = v + (size_t)NTOK * DIMM;
  float* rp = hb + (size_t)NTOK * DIMM;

  const dim3 blk(256);
  const dim3 gG(DIMM / 128, NTOK / 128); // (8, 16) for 2048x1024x1024

  // 1) q,k,v = x @ W^T + b  (fp32 WMMA, async double-buffered LDS)
  gemm_absT<<<gG, blk, 0, stream>>>(x, Wq, bq, q, DIMM, DIMM);
  gemm_absT<<<gG, blk, 0, stream>>>(x, Wk, bk, k, DIMM, DIMM);
  gemm_absT<<<gG, blk, 0, stream>>>(x, Wv, bv, v, DIMM, DIMM);

  // 2) scaled random projections (16 x 64 x 64)
  build_rp<<<dim3(256), blk, 0, stream>>>(log_sigma, rproj, rp);

  // 3) phi features (normalize fused, WMMA, in place); mask on phi_k only
  phi_kernel<<<dim3(NTOK / 128, NH), blk, 0, stream>>>(q, rp, mask, 0);
  phi_kernel<<<dim3(NTOK / 128, NH), blk, 0, stream>>>(k, rp, mask, 1);

  // 4) chunked causal RFA (one WG per (head, batch)); 80.5 KB dynamic LDS
  const size_t rfa_smem = (5 * 4096 + 128) * sizeof(float);
  rfa_kernel<<<dim3(NH, BSZ), blk, rfa_smem, stream>>>(q, k, v, hb);

  // 5) out = h @ Wo^T + bo
  gemm_absT<<<gG, blk, 0, stream>>>(hb, Wo, bo, (float*)d_out, DIMM, DIMM);
}